// DirGNN_model_27642409517071
// MI455X (gfx1250) — compile-verified
//
#include <hip/hip_runtime.h>
#include <hip/hip_bf16.h>

typedef __attribute__((ext_vector_type(2))) float v2f;
typedef __attribute__((ext_vector_type(4))) float v4f;
typedef __attribute__((ext_vector_type(8))) float v8f;

#define ALPHA 0.5f
#define LN_EPS 1e-5f

// ---------------------------------------------------------------------------
// Degree counting (self-loops added analytically as +1 in finish_dinv)
// ---------------------------------------------------------------------------
__global__ void count_deg_kernel(const int* __restrict__ esrc,
                                 const int* __restrict__ edst,
                                 float* __restrict__ deg_d,   // in-degree (dst side)
                                 float* __restrict__ deg_s,   // out-degree (src side)
                                 int n_edges) {
    int e = blockIdx.x * blockDim.x + threadIdx.x;
    if (e >= n_edges) return;
    unsafeAtomicAdd(&deg_d[edst[e]], 1.0f);
    unsafeAtomicAdd(&deg_s[esrc[e]], 1.0f);
}

__global__ void finish_dinv_kernel(float* __restrict__ deg_d,
                                   float* __restrict__ deg_s,
                                   int n) {
    int v = blockIdx.x * blockDim.x + threadIdx.x;
    if (v >= n) return;
    deg_d[v] = rsqrtf(deg_d[v] + 1.0f);   // +1 self-loop; deg>=1 always
    deg_s[v] = rsqrtf(deg_s[v] + 1.0f);
}

// Precompute per-edge norm pair: enorm[e] = {dinv_d[s]*dinv_d[d], dinv_s[s]*dinv_s[d]}
// Edges e >= n_edges are the implicit self-loops (v = e - n_edges).
__global__ void edge_norm_kernel(const int* __restrict__ esrc,
                                 const int* __restrict__ edst,
                                 const float* __restrict__ dinv_d,
                                 const float* __restrict__ dinv_s,
                                 v2f* __restrict__ enorm,
                                 int n_edges, int n_nodes) {
    int e = blockIdx.x * blockDim.x + threadIdx.x;
    if (e >= n_edges + n_nodes) return;
    int s, d;
    if (e < n_edges) { s = esrc[e]; d = edst[e]; }
    else             { s = d = e - n_edges; }
    v2f nm;
    nm.x = dinv_d[s] * dinv_d[d];
    nm.y = dinv_s[s] * dinv_s[d];
    enorm[e] = nm;
}

// ---------------------------------------------------------------------------
// Fused dual GEMM: Out[:,0:64] = H @ Wa ; Out[:,64:128] = H @ Wb
// One wave computes a 16x64 output tile with v_wmma_f32_16x16x4_f32.
// A fragment layout (32-bit A 16x4): lanes 0-15 hold K=0,1 (v0,v1) for M=lane;
// lanes 16-31 hold K=2,3. B mirrors with N in place of M.
// ---------------------------------------------------------------------------
__global__ __launch_bounds__(256)
void gemm64_wmma_kernel(const float* __restrict__ H,   // n x 64
                        const float* __restrict__ Wa,  // 64 x 64
                        const float* __restrict__ Wb,  // 64 x 64
                        float* __restrict__ Out,       // n x 128
                        int nrows) {
    const int wave = threadIdx.x >> 5;
    const int tile = blockIdx.x * 8 + wave;           // 16-row tile index
    const int row0 = tile * 16;
    if (row0 >= nrows) return;                         // wave-uniform exit

    const float* __restrict__ W = blockIdx.y ? Wb : Wa;
    const int outbase = blockIdx.y * 64;

    const int lane = threadIdx.x & 31;
    const int m  = lane & 15;                          // row/col within tile
    const int ks = (lane >> 4) << 1;                   // K sub-offset: 0 or 2

    v8f acc0 = {}, acc1 = {}, acc2 = {}, acc3 = {};

#pragma unroll
    for (int k0 = 0; k0 < 64; k0 += 4) {
        const float* __restrict__ hr = H + (size_t)(row0 + m) * 64 + k0 + ks;
        v2f a; a.x = hr[0]; a.y = hr[1];

        const float* __restrict__ w0 = W + (size_t)(k0 + ks) * 64;
        const float* __restrict__ w1 = w0 + 64;
        v2f b0, b1, b2, b3;
        b0.x = w0[m];      b0.y = w1[m];
        b1.x = w0[16 + m]; b1.y = w1[16 + m];
        b2.x = w0[32 + m]; b2.y = w1[32 + m];
        b3.x = w0[48 + m]; b3.y = w1[48 + m];

        acc0 = __builtin_amdgcn_wmma_f32_16x16x4_f32(false, a, false, b0, (short)0, acc0, false, false);
        acc1 = __builtin_amdgcn_wmma_f32_16x16x4_f32(false, a, false, b1, (short)0, acc1, false, false);
        acc2 = __builtin_amdgcn_wmma_f32_16x16x4_f32(false, a, false, b2, (short)0, acc2, false, false);
        acc3 = __builtin_amdgcn_wmma_f32_16x16x4_f32(false, a, false, b3, (short)0, acc3, false, false);
    }

    // C/D layout: VGPR r, lanes 0-15 -> M=r, N=lane; lanes 16-31 -> M=r+8, N=lane-16
    const int mof = (lane >> 4) << 3;
#pragma unroll
    for (int r = 0; r < 8; ++r) {
        float* __restrict__ orow = Out + (size_t)(row0 + r + mof) * 128 + outbase + m;
        orow[0]  = acc0[r];
        orow[16] = acc1[r];
        orow[32] = acc2[r];
        orow[48] = acc3[r];
    }
}

// ---------------------------------------------------------------------------
// Edge-parallel aggregation, one wave per edge, both directions fused.
// HW[:,0:64] = h@W_in, HW[:,64:128] = h@W_out.
// Lanes 0-15 : in-direction,  gather HW_in[src]  -> scatter agg[dst, 0:64]
// Lanes 16-31: out-direction, gather HW_out[dst] -> scatter agg[src, 64:128]
// Each lane owns a float4 slice -> single global_load_b128 per lane.
// Edges e >= n_edges are the implicit self-loops (v = e - n_edges).
// ---------------------------------------------------------------------------
__global__ __launch_bounds__(256)
void aggregate_kernel(const float* __restrict__ HW,
                      const int* __restrict__ esrc,
                      const int* __restrict__ edst,
                      const v2f* __restrict__ enorm,
                      float* __restrict__ agg,
                      int n_edges, int n_nodes) {
    const int gwave = (blockIdx.x * blockDim.x + threadIdx.x) >> 5;
    if (gwave >= n_edges + n_nodes) return;
    const int lane = threadIdx.x & 31;
    const int half = lane >> 4;                        // 0 = in, 1 = out
    const int c4 = (lane & 15) << 2;                   // float4 column slice

    int s, d;
    if (gwave < n_edges) { s = esrc[gwave]; d = edst[gwave]; }
    else                 { s = d = gwave - n_edges; }

    const v2f nm = enorm[gwave];
    const float nrm = half ? nm.y : nm.x;

    // gather: in-dir reads HW[src, c4..c4+3]; out-dir reads HW[dst, 64+c4..]
    const size_t gsrc = half ? ((size_t)d * 128 + 64) : ((size_t)s * 128);
    const v4f h = *(const v4f*)(HW + gsrc + c4);

    // scatter: in-dir -> agg[dst, c4..]; out-dir -> agg[src, 64+c4..]
    float* __restrict__ p = agg + (half ? ((size_t)s * 128 + 64) : ((size_t)d * 128)) + c4;
    unsafeAtomicAdd(p,     h.x * nrm);
    unsafeAtomicAdd(p + 1, h.y * nrm);
    unsafeAtomicAdd(p + 2, h.z * nrm);
    unsafeAtomicAdd(p + 3, h.w * nrm);
}

// ---------------------------------------------------------------------------
// h_next = maybe_relu( ALPHA*(agg_out + b_out) + (1-ALPHA)*(agg_in + b_in) )
// ---------------------------------------------------------------------------
__global__ void combine_kernel(const float* __restrict__ agg,
                               const float* __restrict__ b_in,
                               const float* __restrict__ b_out,
                               float* __restrict__ hout,
                               int n, int do_relu) {
    int i = blockIdx.x * blockDim.x + threadIdx.x;
    if (i >= n * 64) return;
    int v = i >> 6, c = i & 63;
    float vin  = agg[(size_t)v * 128 + c]      + b_in[c];
    float vout = agg[(size_t)v * 128 + 64 + c] + b_out[c];
    float r = ALPHA * vout + (1.0f - ALPHA) * vin;
    if (do_relu) r = fmaxf(r, 0.0f);
    hout[i] = r;
}

// ---------------------------------------------------------------------------
// Global mean pool (sum + count via atomics)
// ---------------------------------------------------------------------------
__global__ void pool_kernel(const float* __restrict__ h,
                            const int* __restrict__ batch,
                            float* __restrict__ pooled,   // G x 64 sums
                            float* __restrict__ cnt,      // G
                            int n) {
    int i = blockIdx.x * blockDim.x + threadIdx.x;
    if (i >= n * 64) return;
    int v = i >> 6, c = i & 63;
    int g = batch[v];
    unsafeAtomicAdd(&pooled[g * 64 + c], h[i]);
    if (c == 0) unsafeAtomicAdd(&cnt[g], 1.0f);
}

// ---------------------------------------------------------------------------
// Per-graph head: mean, LayerNorm, 64->128 ReLU, 128->2. One block per graph.
// ---------------------------------------------------------------------------
__global__ __launch_bounds__(128)
void head_kernel(const float* __restrict__ pooled,
                 const float* __restrict__ cnt,
                 const float* __restrict__ ln_w,
                 const float* __restrict__ ln_b,
                 const float* __restrict__ P1w,  // 64 x 128
                 const float* __restrict__ P1b,  // 128
                 const float* __restrict__ P2w,  // 128 x 2
                 const float* __restrict__ P2b,  // 2
                 float* __restrict__ out) {      // G x 2
    __shared__ float z[64];
    __shared__ float a1[128];
    __shared__ float stat[2];

    const int g = blockIdx.x;
    const int t = threadIdx.x;
    const float inv_c = 1.0f / fmaxf(cnt[g], 1.0f);

    if (t == 0) {
        float mu = 0.0f;
        for (int i = 0; i < 64; ++i) mu += pooled[g * 64 + i] * inv_c;
        mu *= (1.0f / 64.0f);
        float var = 0.0f;
        for (int i = 0; i < 64; ++i) {
            float dx = pooled[g * 64 + i] * inv_c - mu;
            var += dx * dx;
        }
        var *= (1.0f / 64.0f);
        stat[0] = mu;
        stat[1] = rsqrtf(var + LN_EPS);
    }
    __syncthreads();
    if (t < 64) {
        z[t] = (pooled[g * 64 + t] * inv_c - stat[0]) * stat[1] * ln_w[t] + ln_b[t];
    }
    __syncthreads();

    float s = P1b[t];
    for (int c = 0; c < 64; ++c) s += z[c] * P1w[c * 128 + t];
    a1[t] = fmaxf(s, 0.0f);
    __syncthreads();

    if (t < 2) {
        float s2 = P2b[t];
        for (int j = 0; j < 128; ++j) s2 += a1[j] * P2w[j * 2 + t];
        out[g * 2 + t] = s2;
    }
}

// ---------------------------------------------------------------------------
// Launch
// ---------------------------------------------------------------------------
extern "C" void kernel_launch(void* const* d_in, const int* in_sizes, int n_in,
                              void* d_out, int out_size, void* d_ws, size_t ws_size,
                              hipStream_t stream) {
    const float* x      = (const float*)d_in[0];
    const int*   esrc   = (const int*)  d_in[1];
    const int*   edst   = (const int*)  d_in[2];
    const int*   batch  = (const int*)  d_in[3];
    const float* W1_in  = (const float*)d_in[4];
    const float* b1_in  = (const float*)d_in[5];
    const float* W1_out = (const float*)d_in[6];
    const float* b1_out = (const float*)d_in[7];
    const float* W2_in  = (const float*)d_in[8];
    const float* b2_in  = (const float*)d_in[9];
    const float* W2_out = (const float*)d_in[10];
    const float* b2_out = (const float*)d_in[11];
    const float* W3_in  = (const float*)d_in[12];
    const float* b3_in  = (const float*)d_in[13];
    const float* W3_out = (const float*)d_in[14];
    const float* b3_out = (const float*)d_in[15];
    const float* ln_w   = (const float*)d_in[16];
    const float* ln_b   = (const float*)d_in[17];
    const float* P1w    = (const float*)d_in[18];
    const float* P1b    = (const float*)d_in[19];
    const float* P2w    = (const float*)d_in[20];
    const float* P2b    = (const float*)d_in[21];
    float* out = (float*)d_out;

    const int N = in_sizes[3];        // batch has N entries
    const int E = in_sizes[1];        // edge_src has E entries
    const int G = 64;
    const int EN = E + N;             // edges incl. self-loops

    // Workspace carve (floats)
    float* ws     = (float*)d_ws;
    float* hA     = ws;                          // N x 64
    float* hB     = hA + (size_t)N * 64;         // N x 64
    float* HW     = hB + (size_t)N * 64;         // N x 128
    float* agg    = HW + (size_t)N * 128;        // N x 128
    float* dinv_d = agg + (size_t)N * 128;       // N
    float* dinv_s = dinv_d + N;                  // N
    float* pooled = dinv_s + N;                  // G x 64
    float* cnt    = pooled + (size_t)G * 64;     // G
    v2f*   enorm  = (v2f*)(cnt + G);             // (E+N) x 2

    const int T = 256;
    const int rowTiles = (N + 15) / 16;
    dim3 gemmGrid((rowTiles + 7) / 8, 2);
    const int aggBlocks = ((size_t)EN * 32 + T - 1) / T;
    const int nodeFeatBlocks = ((size_t)N * 64 + T - 1) / T;

    // ---- degrees / normalization (per call; edges are fixed inputs) ----
    hipMemsetAsync(dinv_d, 0, (size_t)2 * N * sizeof(float), stream);
    count_deg_kernel<<<(E + T - 1) / T, T, 0, stream>>>(esrc, edst, dinv_d, dinv_s, E);
    finish_dinv_kernel<<<(N + T - 1) / T, T, 0, stream>>>(dinv_d, dinv_s, N);
    edge_norm_kernel<<<(EN + T - 1) / T, T, 0, stream>>>(esrc, edst, dinv_d, dinv_s, enorm, E, N);

    // ---- layer 1: x -> hA (relu) ----
    gemm64_wmma_kernel<<<gemmGrid, T, 0, stream>>>(x, W1_in, W1_out, HW, N);
    hipMemsetAsync(agg, 0, (size_t)N * 128 * sizeof(float), stream);
    aggregate_kernel<<<aggBlocks, T, 0, stream>>>(HW, esrc, edst, enorm, agg, E, N);
    combine_kernel<<<nodeFeatBlocks, T, 0, stream>>>(agg, b1_in, b1_out, hA, N, 1);

    // ---- layer 2: hA -> hB (relu) ----
    gemm64_wmma_kernel<<<gemmGrid, T, 0, stream>>>(hA, W2_in, W2_out, HW, N);
    hipMemsetAsync(agg, 0, (size_t)N * 128 * sizeof(float), stream);
    aggregate_kernel<<<aggBlocks, T, 0, stream>>>(HW, esrc, edst, enorm, agg, E, N);
    combine_kernel<<<nodeFeatBlocks, T, 0, stream>>>(agg, b2_in, b2_out, hB, N, 1);

    // ---- layer 3: hB -> hA (no relu) ----
    gemm64_wmma_kernel<<<gemmGrid, T, 0, stream>>>(hB, W3_in, W3_out, HW, N);
    hipMemsetAsync(agg, 0, (size_t)N * 128 * sizeof(float), stream);
    aggregate_kernel<<<aggBlocks, T, 0, stream>>>(HW, esrc, edst, enorm, agg, E, N);
    combine_kernel<<<nodeFeatBlocks, T, 0, stream>>>(agg, b3_in, b3_out, hA, N, 0);

    // ---- pool + head ----
    hipMemsetAsync(pooled, 0, (size_t)(G * 64 + G) * sizeof(float), stream);
    pool_kernel<<<nodeFeatBlocks, T, 0, stream>>>(hA, batch, pooled, cnt, N);
    head_kernel<<<G, 128, 0, stream>>>(pooled, cnt, ln_w, ln_b, P1w, P1b, P2w, P2b, out);
}